// YoloLoss_34591666602451
// MI455X (gfx1250) — compile-verified
//
#include <hip/hip_runtime.h>
#include <hip/hip_bf16.h>
#include <math.h>

// YOLO loss, MI455X (gfx1250, wave32).
// Memory-bound: dominant cost is the strided objectness-channel read
// (403,200 cells * 1 cacheline) ~ tens of MB @ 23.3 TB/s. All reductions are
// done with V_WMMA_F32_16X16X4_F32 used as an exact fp32 column-sum engine
// (A = ones(16x4), B = 64 partial sums -> 16 column sums, chained via C).
// Deterministic: fixed block assignment, fixed-order reductions, and the
// tobj scatter resolves duplicate cells with atomicMax on the float bit
// pattern (valid total order for non-negative IoU values).

typedef __attribute__((ext_vector_type(2))) float v2f;
typedef __attribute__((ext_vector_type(8))) float v8f;

#define NCELL0 307200   // 16*3*80*80
#define NCELL1 76800    // 16*3*40*40
#define NCELL2 19200    // 16*3*20*20
#define NCELLS (NCELL0 + NCELL1 + NCELL2)
#define NBLK_C ((NCELLS + 255) / 256)   // 1575, exact

// ---- exact fp32 block reduction of 256 values via WMMA ----------------------
// Returns the block sum (valid on lanes of wave 0; use from threadIdx.x==0).
__device__ __forceinline__ float block_reduce_256(float v, float* red) {
    const int tid = threadIdx.x;
    red[tid] = v;
    __syncthreads();
    float total = 0.0f;
    if (tid < 32) {   // wave 0 only: EXEC all-ones within the wave (uniform branch)
#if __has_builtin(__builtin_amdgcn_wmma_f32_16x16x4_f32)
        v2f a; a.x = 1.0f; a.y = 1.0f;       // A = ones(16x4): D[i][j] = sum_k B[k][j]
        v8f c = {};
        #pragma unroll
        for (int j = 0; j < 4; ++j) {
            v2f b;                            // any bijective packing of 64 vals is fine:
            b.x = red[j * 64 + tid];          // every element lands in exactly one column sum
            b.y = red[j * 64 + 32 + tid];
            c = __builtin_amdgcn_wmma_f32_16x16x4_f32(
                    /*neg_a=*/false, a, /*neg_b=*/false, b,
                    /*c_mod=*/(short)0, c, /*reuse_a=*/false, /*reuse_b=*/false);
        }
        // D row M=0 lives in c[0] of lanes 0..15 (one column sum per lane).
        float d0 = c[0];
        float t = 0.0f;
        #pragma unroll
        for (int j = 0; j < 16; ++j) t += __shfl(d0, j, 32);
        total = t;
#else
        float t = 0.0f;
        for (int j = 0; j < 8; ++j) t += red[j * 32 + tid];
        for (int j = 0; j < 32; ++j) t = (j == 0) ? __shfl(t, 0, 32) + 0.0f : t; // placeholder
        // simple deterministic fallback: lane 0 sums everything
        if (tid == 0) { t = 0.0f; for (int j = 0; j < 256; ++j) t += red[j]; }
        total = t;
#endif
    }
    __syncthreads();   // allow `red` reuse by a subsequent call
    return total;
}

__device__ __forceinline__ float bce_logits(float x, float z) {
    return fmaxf(x, 0.0f) - x * z + log1pf(expf(-fabsf(x)));
}

// ---- per-level target kernel: 1 block per level (600 rows) ------------------
__global__ void yolo_targets_kernel(const float* __restrict__ p0,
                                    const float* __restrict__ p1,
                                    const float* __restrict__ p2,
                                    const float* __restrict__ targets,  // (200,6)
                                    const float* __restrict__ anchors,  // (3,3,2)
                                    float* __restrict__ tobj,           // NCELLS, zeroed
                                    float* __restrict__ lvl_loss) {     // 3 floats
    const int level = blockIdx.x;
    int W, H, toff; const float* p; float stride;
    if (level == 0)      { W = 80; H = 80; p = p0; stride =  8.0f; toff = 0; }
    else if (level == 1) { W = 40; H = 40; p = p1; stride = 16.0f; toff = NCELL0; }
    else                 { W = 20; H = 20; p = p2; stride = 32.0f; toff = NCELL0 + NCELL1; }

    float vf_acc = 0.0f, lbox_acc = 0.0f, lcls_acc = 0.0f;

    for (int row = threadIdx.x; row < 600; row += blockDim.x) {
        const int a  = row / 200;
        const int ti = row - a * 200;
        const float ancx = anchors[(level * 3 + a) * 2 + 0] / stride;
        const float ancy = anchors[(level * 3 + a) * 2 + 1] / stride;
        const float* t = targets + ti * 6;
        const int b   = (int)t[0];
        const int cls = (int)t[1];
        const float tx = t[2] * (float)W, ty = t[3] * (float)H;
        const float tw = t[4] * (float)W, th = t[5] * (float)H;

        // aspect-ratio gate: max(r, 1/r) < 4 on both dims
        const float rx = tw / ancx, ry = th / ancy;
        const float m = fmaxf(fmaxf(rx, 1.0f / rx), fmaxf(ry, 1.0f / ry));
        if (!(m < 4.0f)) continue;           // invalid rows contribute nothing
        vf_acc += 1.0f;

        const float fx = floorf(tx), fy = floorf(ty);
        const int gi = min(max((int)fx, 0), W - 1);
        const int gj = min(max((int)fy, 0), H - 1);

        const size_t cell = ((size_t)(b * 3 + a) * H + gj) * W + gi;
        const float* ps = p + cell * 85;
        __builtin_prefetch(ps, 0, 0);        // global_prefetch_b8: pull the 85-ch row

        const float px = 1.0f / (1.0f + expf(-ps[0]));
        const float py = 1.0f / (1.0f + expf(-ps[1]));
        const float pw = expf(ps[2]) * ancx;
        const float ph = expf(ps[3]) * ancy;

        const float bx = tx - fx, by = ty - fy;  // target box (cell-relative), wh in grid units
        const float b1x1 = px - pw * 0.5f, b1x2 = px + pw * 0.5f;
        const float b1y1 = py - ph * 0.5f, b1y2 = py + ph * 0.5f;
        const float b2x1 = bx - tw * 0.5f, b2x2 = bx + tw * 0.5f;
        const float b2y1 = by - th * 0.5f, b2y2 = by + th * 0.5f;
        const float iw = fmaxf(fminf(b1x2, b2x2) - fmaxf(b1x1, b2x1), 0.0f);
        const float ih = fmaxf(fminf(b1y2, b2y2) - fmaxf(b1y1, b2y1), 0.0f);
        const float inter = iw * ih;
        const float iou = inter / (pw * ph + tw * th - inter + 1e-9f);

        lbox_acc += 1.0f - iou;

        // tobj scatter: deterministic duplicate resolution via bitwise max
        // (non-negative floats order identically as uints; init value is 0).
        atomicMax((unsigned int*)(tobj + toff + cell), __float_as_uint(fmaxf(iou, 0.0f)));

        // class BCE vs one-hot
        float cl = 0.0f;
        #pragma unroll 4
        for (int cc = 0; cc < 80; ++cc)
            cl += bce_logits(ps[5 + cc], (cc == cls) ? 1.0f : 0.0f);
        lcls_acc += cl;
    }

    __shared__ float red[256];
    const float vfs = block_reduce_256(vf_acc,  red);
    const float lbs = block_reduce_256(lbox_acc, red);
    const float lcs = block_reduce_256(lcls_acc, red);
    if (threadIdx.x == 0) {
        const float nv = fmaxf(vfs, 1.0f);
        lvl_loss[level] = lbs / nv + lcs / (nv * 80.0f);
    }
}

// ---- lobj: BCE over every cell's objectness logit vs tobj -------------------
__global__ void yolo_obj_kernel(const float* __restrict__ p0,
                                const float* __restrict__ p1,
                                const float* __restrict__ p2,
                                const float* __restrict__ tobj,
                                float* __restrict__ partials) {
    const int idx = blockIdx.x * blockDim.x + threadIdx.x;
    float acc = 0.0f;
    if (idx < NCELLS) {
        const float* p; int off; float invN;
        if (idx < NCELL0)                { p = p0; off = idx;                   invN = 1.0f / (float)NCELL0; }
        else if (idx < NCELL0 + NCELL1)  { p = p1; off = idx - NCELL0;          invN = 1.0f / (float)NCELL1; }
        else                             { p = p2; off = idx - NCELL0 - NCELL1; invN = 1.0f / (float)NCELL2; }
        const float x = p[(size_t)off * 85 + 4];   // objectness logit (stride-85 gather)
        acc = bce_logits(x, tobj[idx]) * invN;     // pre-scale by 1/mean-count per level
    }
    __shared__ float red[256];
    const float s = block_reduce_256(acc, red);
    if (threadIdx.x == 0) partials[blockIdx.x] = s;
}

// ---- finalize: sum partials (fixed order) + per-level target losses ---------
__global__ void yolo_final_kernel(const float* __restrict__ partials, int n,
                                  const float* __restrict__ lvl_loss,
                                  float* __restrict__ out) {
    float acc = 0.0f;
    for (int i = threadIdx.x; i < n; i += blockDim.x) acc += partials[i];
    __shared__ float red[256];
    const float s = block_reduce_256(acc, red);
    if (threadIdx.x == 0)
        out[0] = s + lvl_loss[0] + lvl_loss[1] + lvl_loss[2];
}

extern "C" void kernel_launch(void* const* d_in, const int* in_sizes, int n_in,
                              void* d_out, int out_size, void* d_ws, size_t ws_size,
                              hipStream_t stream) {
    const float* p0      = (const float*)d_in[0];  // (16,3,80,80,85)
    const float* p1      = (const float*)d_in[1];  // (16,3,40,40,85)
    const float* p2      = (const float*)d_in[2];  // (16,3,20,20,85)
    const float* targets = (const float*)d_in[3];  // (200,6)
    const float* anchors = (const float*)d_in[4];  // (3,3,2)

    float* ws       = (float*)d_ws;
    float* tobj     = ws;                 // NCELLS floats
    float* lvl_loss = ws + NCELLS;        // 3 floats
    float* partials = ws + NCELLS + 3;    // NBLK_C floats

    // tobj must be zero every call (harness does not re-poison/zero).
    hipMemsetAsync(tobj, 0, (size_t)NCELLS * sizeof(float), stream);

    yolo_targets_kernel<<<3, 256, 0, stream>>>(p0, p1, p2, targets, anchors,
                                               tobj, lvl_loss);
    yolo_obj_kernel<<<NBLK_C, 256, 0, stream>>>(p0, p1, p2, tobj, partials);
    yolo_final_kernel<<<1, 256, 0, stream>>>(partials, NBLK_C, lvl_loss,
                                             (float*)d_out);
}